// SpatitalAttention_90769838833934
// MI455X (gfx1250) — compile-verified
//
#include <hip/hip_runtime.h>

// ---------------------------------------------------------------------------
// Problem constants (reference: B=2, C=64, Z=H=W=64, WIN=9, pad=4 SAME)
// ---------------------------------------------------------------------------
#define BATCH   2
#define CDIM    64
#define NVOX    (64 * 64 * 64)        // 262144 voxels per (b, c)
#define NTILES  (NVOX / 16)           // 16384 16-column tiles per batch
#define TOTAL_ELEMS (2LL * 64 * 64 * 64 * 64)  // 33,554,432 per tensor

typedef float v2f __attribute__((ext_vector_type(2)));
typedef float v8f __attribute__((ext_vector_type(8)));

#if defined(__has_builtin)
#if __has_builtin(__builtin_amdgcn_tensor_load_to_lds) && \
    __has_builtin(__builtin_amdgcn_s_wait_tensorcnt)
#define HAVE_TDM 1
#endif
#endif
#ifndef HAVE_TDM
#define HAVE_TDM 0
#endif

// ---------------------------------------------------------------------------
// Stage 1: fused QKV projection (V_WMMA_F32_16X16X4_F32) + qk / qkv products.
//   q[l,n] = sum_c q_w[l,c] * moving[c,n]   (per batch), same for k,v on fixed
//   qk = q*k ; qkv = qk*v  -> streamed to workspace, q/k/v never hit memory.
// One wave owns a 64(l) x 16(n) tile: 4 l-subtiles x 16 K-steps x 3 GEMMs.
// ---------------------------------------------------------------------------
__global__ __launch_bounds__(128)
void qkv_proj_kernel(const float* __restrict__ moving,
                     const float* __restrict__ fixedp,
                     const float* __restrict__ qw,
                     const float* __restrict__ kw,
                     const float* __restrict__ vw,
                     float* __restrict__ qk_out,
                     float* __restrict__ qkv_out)
{
    const int wave = threadIdx.x >> 5;           // 0..3
    const int lane = threadIdx.x & 31;
    const long long tile = (long long)blockIdx.x * 4 + wave;  // global n-tile
    const int b  = (int)(tile / NTILES);
    const int nt = (int)(tile % NTILES);
    const int n0 = nt * 16;

    const float* __restrict__ M = moving + (long long)b * CDIM * NVOX;
    const float* __restrict__ F = fixedp + (long long)b * CDIM * NVOX;

    // WMMA f32 16x16x4 operand addressing (per CDNA5 ISA):
    //  A (16x4): lanes 0-15 -> M rows 0-15 / K = {0,1}; lanes 16-31 -> K = {2,3}
    //  B (4x16): lanes hold N columns; VGPR pair holds K = {khalf, khalf+1}
    const int mrow  = lane & 15;
    const int khalf = (lane >> 4) << 1;          // 0 or 2
    const int ncol  = n0 + (lane & 15);

    v8f qacc[4] = {};
    v8f kacc[4] = {};
    v8f vacc[4] = {};

    for (int kk = 0; kk < 16; ++kk) {
        const int kbase = kk * 4 + khalf;

        // B fragments for this K-step: one from moving (for q), one from fixed
        v2f bm, bf;
        bm.x = M[(long long)kbase * NVOX + ncol];
        bm.y = M[(long long)(kbase + 1) * NVOX + ncol];
        bf.x = F[(long long)kbase * NVOX + ncol];
        bf.y = F[(long long)(kbase + 1) * NVOX + ncol];

#pragma unroll
        for (int lt = 0; lt < 4; ++lt) {
            const int row = lt * 16 + mrow;
            // 8-byte aligned pair loads of weights (kbase is even)
            v2f aq = *(const v2f*)(qw + row * 64 + kbase);
            v2f ak = *(const v2f*)(kw + row * 64 + kbase);
            v2f av = *(const v2f*)(vw + row * 64 + kbase);

            qacc[lt] = __builtin_amdgcn_wmma_f32_16x16x4_f32(
                false, aq, false, bm, (short)0, qacc[lt], false, false);
            kacc[lt] = __builtin_amdgcn_wmma_f32_16x16x4_f32(
                false, ak, false, bf, (short)0, kacc[lt], false, false);
            vacc[lt] = __builtin_amdgcn_wmma_f32_16x16x4_f32(
                false, av, false, bf, (short)0, vacc[lt], false, false);
        }
    }

    // C/D layout: VGPR r holds M = r (lanes 0-15) or M = r+8 (lanes 16-31),
    // N = lane & 15.  Form qk / qkv in registers and stream out.
    float* __restrict__ QK  = qk_out  + (long long)b * CDIM * NVOX;
    float* __restrict__ QKV = qkv_out + (long long)b * CDIM * NVOX;
    const int rowoff = (lane >> 4) << 3;         // +8 for upper half-wave

#pragma unroll
    for (int lt = 0; lt < 4; ++lt) {
#pragma unroll
        for (int r = 0; r < 8; ++r) {
            const float qkv_qk = qacc[lt][r] * kacc[lt][r];
            const float qkv_v  = qkv_qk * vacc[lt][r];
            const int   l      = lt * 16 + r + rowoff;
            const long long idx = (long long)l * NVOX + n0 + (lane & 15);
            QK[idx]  = qkv_qk;
            QKV[idx] = qkv_v;
        }
    }
}

#if HAVE_TDM
// ---------------------------------------------------------------------------
// TDM helper: DMA one contiguous 4096-float (16 KB) z-slice into LDS.
// D# per CDNA5 ISA:
//   group0: count=1 | lds_addr | global_addr[56:0] | type=2
//   group1: data_size=4B, tensor_dim0=tile_dim0=4096, tensor_dim1=tile_dim1=1
//   groups 2/3 (+ extra group on this toolchain): zero (2-D descriptor)
// amdgpu-toolchain (clang-23) builtin arity: 6 args
//   (u32x4 g0, i32x8 g1, i32x4 g2, i32x4 g3, i32x8 extra, i32 cpol)
// ---------------------------------------------------------------------------
typedef unsigned int tdm_u4 __attribute__((ext_vector_type(4)));
typedef int          tdm_i8 __attribute__((ext_vector_type(8)));
typedef int          tdm_i4 __attribute__((ext_vector_type(4)));

__device__ __forceinline__ void tdm_load_slice16k(const float* gsrc,
                                                  unsigned ldsByteOff)
{
    const unsigned long long ga = (unsigned long long)(const void*)gsrc;
    tdm_u4 g0;
    g0.x = 1u;                                            // count=1 (user D#)
    g0.y = ldsByteOff;                                    // lds_addr (bytes)
    g0.z = (unsigned)ga;                                  // global_addr[31:0]
    g0.w = (unsigned)((ga >> 32) & 0x01FFFFFFull) | (2u << 30); // [56:32]|type=2
    tdm_i8 g1;
    g1[0] = (int)(2u << 16);        // workgroup_mask=0, data_size=2 (4 bytes)
    g1[1] = (int)(4096u << 16);     // tensor_dim0[15:0] = 4096
    g1[2] = (int)(1u << 16);        // tensor_dim0[31:16]=0 | tensor_dim1[15:0]=1
    g1[3] = (int)(4096u << 16);     // tensor_dim1[31:16]=0 | tile_dim0=4096
    g1[4] = 1;                      // tile_dim1=1, tile_dim2=0
    g1[5] = 4096;                   // tensor_dim0_stride[31:0]
    g1[6] = 0;
    g1[7] = 0;
    const tdm_i4 gz4 = {0, 0, 0, 0};
    const tdm_i8 gz8 = {0, 0, 0, 0, 0, 0, 0, 0};
    __builtin_amdgcn_tensor_load_to_lds(g0, g1, gz4, gz4, gz8, 0);
}
#endif // HAVE_TDM

// ---------------------------------------------------------------------------
// Stage 2: fully fused separable 9x9x9 box sum + divide.
// One block owns half of one (b,c) 64^3 volume (32 output z-slices).
// z-slices stream through a DOUBLE-BUFFERED LDS stage filled by the Tensor
// Data Mover (slice z+1 DMAs while slice z is filtered), then W-pass (LDS) ->
// H-pass (regs) -> 9-deep register ring of z-window accumulators
// (each of 1024 threads owns 4 voxels -> ring = 9*4*2 = 72 VGPRs).
// Output slice zo emits qkv_sum/qk_sum once its input window closes.
// The 40-iteration z loop is fully unrolled so ring indices stay static and
// the accumulators live in VGPRs.
// ---------------------------------------------------------------------------
__global__ __launch_bounds__(1024)
void box3d_div_kernel(const float* __restrict__ qk,
                      const float* __restrict__ qkv,
                      float* __restrict__ out)
{
    __shared__ float rawS[2][2][64 * 64]; // [buffer][tensor] raw slice (64 KB)
    __shared__ float wfS [2][64 * 64];    // W-filtered scratch        (32 KB)

    const int bc    = blockIdx.x >> 1;   // 0..127  (= b*C + c)
    const int chunk = blockIdx.x & 1;    // z-half
    const int o0    = chunk * 32;        // outputs zo in [o0, o0+31]
    const int zlo   = (o0 - 4 < 0) ? 0 : o0 - 4;      // first input slice
    const int zhi   = (o0 + 35 > 63) ? 63 : o0 + 35;  // last input slice

    const int t  = threadIdx.x;
    const int h  = t >> 4;               // 0..63
    const int w4 = (t & 15) << 2;        // 0,4,...,60  (owns 4 voxels)
    const int hw = h * 64 + w4;

    const float* __restrict__ qkB  = qk  + (long long)bc * NVOX;
    const float* __restrict__ qkvB = qkv + (long long)bc * NVOX;
    float* __restrict__ outB = out + (long long)bc * NVOX;

    float accA[9][4];                    // ring: windowed qk sums
    float accB[9][4];                    // ring: windowed qkv sums
#pragma unroll
    for (int s = 0; s < 9; ++s)
#pragma unroll
        for (int j = 0; j < 4; ++j) { accA[s][j] = 0.0f; accB[s][j] = 0.0f; }

#pragma unroll
    for (int i = 0; i < 40; ++i) {
        const int z = o0 - 4 + i;        // input slice for this iteration
        if (z >= 0 && z <= 63) {         // block-uniform predicate
            const int cur = (z - zlo) & 1;

#if HAVE_TDM
            // Prologue: first slice of this block's stream.
            if (z == zlo && t == 0) {
                tdm_load_slice16k(qkB + (long long)z * 4096,
                    (unsigned)(unsigned long long)(const void*)&rawS[cur][0][0]);
                tdm_load_slice16k(qkvB + (long long)z * 4096,
                    (unsigned)(unsigned long long)(const void*)&rawS[cur][1][0]);
            }
            __syncthreads();   // everyone done reading buffer cur^1 (iter i-1)
            if (t == 0) {
                if (z < zhi) {
                    // kick DMA of slice z+1 into the other buffer, then wait
                    // for the 2 in-flight ops of the CURRENT slice (in-order).
                    tdm_load_slice16k(qkB + (long long)(z + 1) * 4096,
                        (unsigned)(unsigned long long)(const void*)&rawS[cur ^ 1][0][0]);
                    tdm_load_slice16k(qkvB + (long long)(z + 1) * 4096,
                        (unsigned)(unsigned long long)(const void*)&rawS[cur ^ 1][1][0]);
                    __builtin_amdgcn_s_wait_tensorcnt(2);
                } else {
                    __builtin_amdgcn_s_wait_tensorcnt(0);
                }
            }
            __syncthreads();   // current slice resident in LDS for all waves
#else
            const long long sOff = (long long)z * 4096 + hw;
            const float4 rq = *(const float4*)(qkB  + sOff);
            const float4 rv = *(const float4*)(qkvB + sOff);
            __syncthreads();
            *(float4*)(&rawS[cur][0][hw]) = rq;
            *(float4*)(&rawS[cur][1][hw]) = rv;
            __syncthreads();
#endif

            // ---- W pass (along fastest axis) ----
            float wq[4] = {0, 0, 0, 0}, wv[4] = {0, 0, 0, 0};
#pragma unroll
            for (int dw = -4; dw <= 4; ++dw) {
#pragma unroll
                for (int j = 0; j < 4; ++j) {
                    const int w = w4 + j + dw;
                    if (w >= 0 && w < 64) {
                        wq[j] += rawS[cur][0][h * 64 + w];
                        wv[j] += rawS[cur][1][h * 64 + w];
                    }
                }
            }
#pragma unroll
            for (int j = 0; j < 4; ++j) {
                wfS[0][hw + j] = wq[j];
                wfS[1][hw + j] = wv[j];
            }
            __syncthreads();

            // ---- H pass into registers ----
            float whq[4] = {0, 0, 0, 0}, whv[4] = {0, 0, 0, 0};
#pragma unroll
            for (int dh = -4; dh <= 4; ++dh) {
                const int hh = h + dh;
                if (hh >= 0 && hh < 64) {
#pragma unroll
                    for (int j = 0; j < 4; ++j) {
                        whq[j] += wfS[0][hh * 64 + w4 + j];
                        whv[j] += wfS[1][hh * 64 + w4 + j];
                    }
                }
            }

            // ---- scatter WH(z) into pending z-window accumulators ----
            // zo - o0 = (i - 4) + dz, valid when in [0, 31]; slot static.
#pragma unroll
            for (int dz = -4; dz <= 4; ++dz) {
                const int oi = i - 4 + dz;
                if (oi >= 0 && oi <= 31) {
                    const int s = oi % 9;
#pragma unroll
                    for (int j = 0; j < 4; ++j) {
                        accA[s][j] += whq[j];
                        accB[s][j] += whv[j];
                    }
                }
            }
        }

        // ---- emit output zo = o0 + (i-8): its window closed this iter ----
        if (i >= 8) {                    // compile-time per unrolled i
            const int oi = i - 8;        // 0..31
            const int s  = oi % 9;
            float4 r;
            r.x = accB[s][0] / accA[s][0];
            r.y = accB[s][1] / accA[s][1];
            r.z = accB[s][2] / accA[s][2];
            r.w = accB[s][3] / accA[s][3];
            *(float4*)(outB + (long long)(o0 + oi) * 4096 + hw) = r;
#pragma unroll
            for (int j = 0; j < 4; ++j) { accA[s][j] = 0.0f; accB[s][j] = 0.0f; }
        }
    }
}

// ---------------------------------------------------------------------------
// Launcher.  Inputs (setup_inputs order): moving, fixed, q_w, k_w, v_w (f32).
// Workspace: qk + qkv tensors (128 MB each).
// ---------------------------------------------------------------------------
extern "C" void kernel_launch(void* const* d_in, const int* in_sizes, int n_in,
                              void* d_out, int out_size, void* d_ws, size_t ws_size,
                              hipStream_t stream)
{
    (void)in_sizes; (void)n_in; (void)out_size; (void)ws_size;

    const float* moving = (const float*)d_in[0];
    const float* fixedp = (const float*)d_in[1];
    const float* qw     = (const float*)d_in[2];
    const float* kw     = (const float*)d_in[3];
    const float* vw     = (const float*)d_in[4];
    float* out = (float*)d_out;

    float* ws0 = (float*)d_ws;                   // qk
    float* ws1 = ws0 + TOTAL_ELEMS;              // qkv

    // Stage 1: 32768 n-tiles total, 4 waves (tiles) per 128-thread block
    const int projBlocks = (BATCH * NTILES) / 4; // 8192
    qkv_proj_kernel<<<projBlocks, 128, 0, stream>>>(moving, fixedp, qw, kw, vw,
                                                    ws0, ws1);

    // Stage 2: fused 9x9x9 box sum + divide with TDM-fed double-buffered LDS.
    // 2 blocks per (b,c) volume (one per 32-slice z-half), 1024 threads each.
    box3d_div_kernel<<<BATCH * CDIM * 2, 1024, 0, stream>>>(ws0, ws1, out);
}